// Transformer_77318001262622
// MI455X (gfx1250) — compile-verified
//
#include <hip/hip_runtime.h>
#include <math.h>

// ---------------------------------------------------------------------------
// Transformer encoder-decoder forward for MI455X (gfx1250), fp32 end-to-end.
// Matrix ops use V_WMMA_F32_16X16X4_F32 (exact fp32 on the WMMA pipe).
// Attention is fused (scores in LDS, per-wave softmax) to avoid a 67MB
// score tensor. All launches on `stream`; no allocation / sync (graph-safe).
// Workspace: 13 * B*S*D floats = 104 MB.
// ---------------------------------------------------------------------------

namespace {

constexpr int kB  = 4;
constexpr int kS  = 512;     // S_SRC == S_TGT
constexpr int kV  = 32000;
constexpr int kNL = 6;
constexpr int kD  = 1024;
constexpr int kF  = 4096;
constexpr int kH  = 16;
constexpr int kDH = 64;
constexpr int kSROW = 516;           // padded LDS score-row stride (floats)
constexpr float kEmbScale = 32.0f;   // sqrt(D)
constexpr float kQScale   = 0.125f;  // 1/sqrt(DH)

typedef float v2f __attribute__((ext_vector_type(2)));
typedef float v8f __attribute__((ext_vector_type(8)));

__device__ inline v8f wmma4(v2f a, v2f b, v8f c) {
  // D = A(16x4, f32) * B(4x16, f32) + C(16x16, f32)
  return __builtin_amdgcn_wmma_f32_16x16x4_f32(false, a, false, b,
                                               (short)0, c, false, false);
}

__device__ inline v8f v8f_zero() {
  v8f z = {0.f, 0.f, 0.f, 0.f, 0.f, 0.f, 0.f, 0.f};
  return z;
}

// ---------------- embedding + positional encoding ----------------
__global__ __launch_bounds__(256)
void embed_kernel(const int* __restrict__ ids, const float* __restrict__ emb,
                  float* __restrict__ out) {
  int row = blockIdx.x;            // b*kS + s
  int s = row & (kS - 1);
  int id = ids[row];
  const float* er = emb + (size_t)id * kD;
  float* orow = out + (size_t)row * kD;
  for (int d = threadIdx.x; d < kD; d += 256) {
    int c = (d < kD / 2) ? d : (d - kD / 2);
    float inv = powf(10000.0f, -2.0f * (float)c / (float)kD);
    float ang = (float)s * inv;
    float pe = (d < kD / 2) ? sinf(ang) : cosf(ang);
    orow[d] = er[d] * kEmbScale + pe;
  }
}

// ---------------- layer norm ----------------
__device__ inline float block_sum(float v, float* sbuf) {
#pragma unroll
  for (int off = 16; off > 0; off >>= 1) v += __shfl_xor(v, off, 32);
  int wave = threadIdx.x >> 5, lane = threadIdx.x & 31;
  if (lane == 0) sbuf[wave] = v;
  __syncthreads();
  if (threadIdx.x < 8) {
    float r = sbuf[threadIdx.x];
#pragma unroll
    for (int off = 4; off > 0; off >>= 1) r += __shfl_xor(r, off, 32);
    if (threadIdx.x == 0) sbuf[0] = r;
  }
  __syncthreads();
  float res = sbuf[0];
  __syncthreads();
  return res;
}

__global__ __launch_bounds__(256)
void ln_kernel(const float* __restrict__ x, const float* __restrict__ g,
               const float* __restrict__ b, float* __restrict__ out) {
  __shared__ float sbuf[8];
  size_t row = blockIdx.x;
  const float* xr = x + row * kD;
  float s = 0.f;
  for (int d = threadIdx.x; d < kD; d += 256) s += xr[d];
  float mean = block_sum(s, sbuf) * (1.0f / kD);
  float vv = 0.f;
  for (int d = threadIdx.x; d < kD; d += 256) {
    float t = xr[d] - mean;
    vv += t * t;
  }
  float var = block_sum(vv, sbuf) * (1.0f / kD);
  float inv = rsqrtf(var + 1e-5f);
  float* orow = out + row * kD;
  for (int d = threadIdx.x; d < kD; d += 256)
    orow[d] = (xr[d] - mean) * inv * g[d] + b[d];
}

// ---------------- WMMA GEMM: C = A[MxK] @ W[KxN] (+bias)(+res)(ReLU) -------
// Block: 256 threads = 8 waves; tile 64(M) x 128(N); wave tile 16 x 64.
// BT=true reads W as [N][K] (i.e. multiplies by W^T) — used for logits.
template <bool BT, bool BIAS, bool RES, bool RELU>
__global__ __launch_bounds__(256)
void gemm_kernel(const float* __restrict__ A, const float* __restrict__ W,
                 const float* __restrict__ bias, const float* res,
                 float* Cout, int M, int N, int K) {
  int wave = threadIdx.x >> 5, lane = threadIdx.x & 31;
  int lo = lane & 15, hi = lane >> 4;
  int m0 = blockIdx.x * 64 + (wave & 3) * 16;
  int n0 = blockIdx.y * 128 + (wave >> 2) * 64;

  v8f c0 = v8f_zero(), c1 = c0, c2 = c0, c3 = c0;
  const float* arow = A + (size_t)(m0 + lo) * K + 2 * hi;
  for (int k0 = 0; k0 < K; k0 += 4) {
    v2f a = *(const v2f*)(arow + k0);
    v2f b0, b1, b2, b3;
    if (BT) {
      const float* wc = W + (size_t)(n0 + lo) * K + k0 + 2 * hi;
      b0 = *(const v2f*)(wc);
      b1 = *(const v2f*)(wc + (size_t)16 * K);
      b2 = *(const v2f*)(wc + (size_t)32 * K);
      b3 = *(const v2f*)(wc + (size_t)48 * K);
    } else {
      const float* wr0 = W + (size_t)(k0 + 2 * hi) * N + n0 + lo;
      const float* wr1 = wr0 + N;
      b0.x = wr0[0];  b0.y = wr1[0];
      b1.x = wr0[16]; b1.y = wr1[16];
      b2.x = wr0[32]; b2.y = wr1[32];
      b3.x = wr0[48]; b3.y = wr1[48];
    }
    c0 = wmma4(a, b0, c0);
    c1 = wmma4(a, b1, c1);
    c2 = wmma4(a, b2, c2);
    c3 = wmma4(a, b3, c3);
  }
  v8f acc[4] = {c0, c1, c2, c3};
#pragma unroll
  for (int t = 0; t < 4; ++t) {
    int n = n0 + t * 16 + lo;
    float bv = BIAS ? bias[n] : 0.0f;
#pragma unroll
    for (int j = 0; j < 8; ++j) {
      int m = m0 + j + 8 * hi;
      float v = acc[t][j] + bv;
      if (RES) v += res[(size_t)m * N + n];
      if (RELU) v = fmaxf(v, 0.0f);
      Cout[(size_t)m * N + n] = v;
    }
  }
}

// ---------------- fused attention ----------------
// Grid: (S/64, H, B); block 128 = 4 waves; each wave owns 16 query rows.
// Phase 1: scores = (Q*scale) @ K^T + mask -> LDS (16 x 512 per wave)
// Phase 2: per-row softmax in LDS (lane pair per row, shfl across halves)
// Phase 3: out = P @ V via WMMA; write merged-head layout.
// MODE 0: key-padding mask only.  MODE 1: causal + key-padding.
template <int MODE>
__global__ __launch_bounds__(128)
void attn_kernel(const float* __restrict__ Qb, int ldq,
                 const float* __restrict__ Kb, int ldk,
                 const float* __restrict__ Vb, int ldv,
                 const int* __restrict__ ids, float* __restrict__ Ob, int ldo) {
  extern __shared__ float smem[];
  int wave = threadIdx.x >> 5, lane = threadIdx.x & 31;
  int lo = lane & 15, hi = lane >> 4;
  int bb = blockIdx.z, hh = blockIdx.y;
  const float* Q  = Qb + (size_t)bb * kS * ldq + hh * kDH;
  const float* Kp = Kb + (size_t)bb * kS * ldk + hh * kDH;
  const float* Vp = Vb + (size_t)bb * kS * ldv + hh * kDH;
  float* O = Ob + (size_t)bb * kS * ldo + hh * kDH;
  const int* kid = ids + bb * kS;
  int q0 = blockIdx.x * 64 + wave * 16;
  float* Sw = smem + wave * (16 * kSROW);

  // ---- Phase 1: masked scores into LDS ----
  v2f qa[16];
  {
    const float* qrow = Q + (size_t)(q0 + lo) * ldq + 2 * hi;
#pragma unroll
    for (int kk = 0; kk < 16; ++kk) {
      v2f t = *(const v2f*)(qrow + kk * 4);
      t.x *= kQScale;
      t.y *= kQScale;
      qa[kk] = t;
    }
  }
  for (int kt = 0; kt < kS / 16; ++kt) {
    v8f s = v8f_zero();
    const float* krow = Kp + (size_t)(kt * 16 + lo) * ldk + 2 * hi;
#pragma unroll
    for (int kk = 0; kk < 16; ++kk) {
      v2f b = *(const v2f*)(krow + kk * 4);
      s = wmma4(qa[kk], b, s);
    }
    int key = kt * 16 + lo;
    float km = (kid[key] == 0) ? -INFINITY : 0.0f;
#pragma unroll
    for (int j = 0; j < 8; ++j) {
      float v = s[j] + km;
      if (MODE == 1 && key > q0 + j + 8 * hi) v = -INFINITY;
      Sw[(j + 8 * hi) * kSROW + key] = v;
    }
  }

  // ---- Phase 2: softmax over kS keys (row = lane%16, half = lane/16) ----
  {
    float* prow = Sw + lo * kSROW + hi * (kS / 2);
    float mx = -INFINITY;
    for (int i = 0; i < kS / 2; i += 4) {
      float4 v = *(float4*)(prow + i);
      mx = fmaxf(mx, fmaxf(fmaxf(v.x, v.y), fmaxf(v.z, v.w)));
    }
    mx = fmaxf(mx, __shfl_xor(mx, 16, 32));
    float sum = 0.f;
    for (int i = 0; i < kS / 2; i += 4) {
      float4 v = *(float4*)(prow + i);
      v.x = __expf(v.x - mx); v.y = __expf(v.y - mx);
      v.z = __expf(v.z - mx); v.w = __expf(v.w - mx);
      *(float4*)(prow + i) = v;
      sum += v.x + v.y + v.z + v.w;
    }
    sum += __shfl_xor(sum, 16, 32);
    float inv = 1.0f / sum;
    for (int i = 0; i < kS / 2; i += 4) {
      float4 v = *(float4*)(prow + i);
      v.x *= inv; v.y *= inv; v.z *= inv; v.w *= inv;
      *(float4*)(prow + i) = v;
    }
  }

  // ---- Phase 3: out = P @ V ----
  v8f o0 = v8f_zero(), o1 = o0, o2 = o0, o3 = o0;
  for (int k0 = 0; k0 < kS; k0 += 4) {
    v2f a = *(const v2f*)(Sw + lo * kSROW + k0 + 2 * hi);
    const float* vr0 = Vp + (size_t)(k0 + 2 * hi) * ldv;
    const float* vr1 = vr0 + ldv;
    v2f b0, b1, b2, b3;
    b0.x = vr0[lo];      b0.y = vr1[lo];
    b1.x = vr0[16 + lo]; b1.y = vr1[16 + lo];
    b2.x = vr0[32 + lo]; b2.y = vr1[32 + lo];
    b3.x = vr0[48 + lo]; b3.y = vr1[48 + lo];
    o0 = wmma4(a, b0, o0);
    o1 = wmma4(a, b1, o1);
    o2 = wmma4(a, b2, o2);
    o3 = wmma4(a, b3, o3);
  }
  v8f oc[4] = {o0, o1, o2, o3};
#pragma unroll
  for (int t = 0; t < 4; ++t)
#pragma unroll
    for (int j = 0; j < 8; ++j)
      O[(size_t)(q0 + j + 8 * hi) * ldo + t * 16 + lo] = oc[t][j];
}

}  // namespace

// ---------------------------------------------------------------------------
extern "C" void kernel_launch(void* const* d_in, const int* in_sizes, int n_in,
                              void* d_out, int out_size, void* d_ws,
                              size_t ws_size, hipStream_t stream) {
  (void)in_sizes; (void)n_in; (void)out_size; (void)ws_size;

  const int*   src_ids = (const int*)d_in[0];
  const int*   tgt_ids = (const int*)d_in[1];
  const float* src_emb = (const float*)d_in[2];
  const float* tgt_emb = (const float*)d_in[3];
  const float* e_ln1_g = (const float*)d_in[4];
  const float* e_ln1_b = (const float*)d_in[5];
  const float* e_wqkv  = (const float*)d_in[6];
  const float* e_bqkv  = (const float*)d_in[7];
  const float* e_wo    = (const float*)d_in[8];
  const float* e_bo    = (const float*)d_in[9];
  const float* e_ln2_g = (const float*)d_in[10];
  const float* e_ln2_b = (const float*)d_in[11];
  const float* e_w1    = (const float*)d_in[12];
  const float* e_b1    = (const float*)d_in[13];
  const float* e_w2    = (const float*)d_in[14];
  const float* e_b2    = (const float*)d_in[15];
  const float* ef_g    = (const float*)d_in[16];
  const float* ef_b    = (const float*)d_in[17];
  const float* t_ln1_g = (const float*)d_in[18];
  const float* t_ln1_b = (const float*)d_in[19];
  const float* t_wqkv  = (const float*)d_in[20];
  const float* t_bqkv  = (const float*)d_in[21];
  const float* t_wo1   = (const float*)d_in[22];
  const float* t_bo1   = (const float*)d_in[23];
  const float* t_ln2_g = (const float*)d_in[24];
  const float* t_ln2_b = (const float*)d_in[25];
  const float* t_wq    = (const float*)d_in[26];
  const float* t_bq    = (const float*)d_in[27];
  const float* t_wkv   = (const float*)d_in[28];
  const float* t_bkv   = (const float*)d_in[29];
  const float* t_wo2   = (const float*)d_in[30];
  const float* t_bo2   = (const float*)d_in[31];
  const float* t_ln3_g = (const float*)d_in[32];
  const float* t_ln3_b = (const float*)d_in[33];
  const float* t_w1    = (const float*)d_in[34];
  const float* t_b1    = (const float*)d_in[35];
  const float* t_w2    = (const float*)d_in[36];
  const float* t_b2    = (const float*)d_in[37];
  const float* df_g    = (const float*)d_in[38];
  const float* df_b    = (const float*)d_in[39];

  const int M = kB * kS;  // 2048 token rows
  float* ws = (float*)d_ws;
  size_t C1 = (size_t)M * kD;   // 2M floats
  float* xs   = ws;             // encoder stream
  float* xt   = xs + C1;        // decoder stream
  float* hbuf = xt + C1;        // layernorm output
  float* qkv  = hbuf + C1;      // [M, 3D] (also reused for cross KV [M, 2D])
  float* ctx  = qkv + 3 * C1;   // merged attention output [M, D]
  float* mid  = ctx + C1;       // FFN hidden [M, F]
  float* qb   = mid + 4 * C1;   // cross-attention Q [M, D]
  float* memf = qb + C1;        // encoder memory (post final LN) [M, D]

  const size_t ATTN_LDS = (size_t)4 * 16 * kSROW * sizeof(float);  // 129 KB
  (void)hipFuncSetAttribute((const void*)attn_kernel<0>,
                            hipFuncAttributeMaxDynamicSharedMemorySize,
                            (int)ATTN_LDS);
  (void)hipFuncSetAttribute((const void*)attn_kernel<1>,
                            hipFuncAttributeMaxDynamicSharedMemorySize,
                            (int)ATTN_LDS);

  dim3 blk256(256), blk128(128);
  dim3 attn_grid(kS / 64, kH, kB);

  // ---- embeddings + positional encodings ----
  embed_kernel<<<M, blk256, 0, stream>>>(src_ids, src_emb, xs);
  embed_kernel<<<M, blk256, 0, stream>>>(tgt_ids, tgt_emb, xt);

  // ---- encoder ----
  for (int l = 0; l < kNL; ++l) {
    ln_kernel<<<M, blk256, 0, stream>>>(xs, e_ln1_g + l * kD, e_ln1_b + l * kD, hbuf);
    gemm_kernel<false, true, false, false><<<dim3(M / 64, 3 * kD / 128), blk256, 0, stream>>>(
        hbuf, e_wqkv + (size_t)l * kD * 3 * kD, e_bqkv + (size_t)l * 3 * kD,
        nullptr, qkv, M, 3 * kD, kD);
    attn_kernel<0><<<attn_grid, blk128, ATTN_LDS, stream>>>(
        qkv, 3 * kD, qkv + kD, 3 * kD, qkv + 2 * kD, 3 * kD, src_ids, ctx, kD);
    gemm_kernel<false, true, true, false><<<dim3(M / 64, kD / 128), blk256, 0, stream>>>(
        ctx, e_wo + (size_t)l * kD * kD, e_bo + (size_t)l * kD, xs, xs, M, kD, kD);
    ln_kernel<<<M, blk256, 0, stream>>>(xs, e_ln2_g + l * kD, e_ln2_b + l * kD, hbuf);
    gemm_kernel<false, true, false, true><<<dim3(M / 64, kF / 128), blk256, 0, stream>>>(
        hbuf, e_w1 + (size_t)l * kD * kF, e_b1 + (size_t)l * kF, nullptr, mid, M, kF, kD);
    gemm_kernel<false, true, true, false><<<dim3(M / 64, kD / 128), blk256, 0, stream>>>(
        mid, e_w2 + (size_t)l * kF * kD, e_b2 + (size_t)l * kD, xs, xs, M, kD, kF);
  }
  ln_kernel<<<M, blk256, 0, stream>>>(xs, ef_g, ef_b, memf);

  // ---- decoder ----
  for (int l = 0; l < kNL; ++l) {
    ln_kernel<<<M, blk256, 0, stream>>>(xt, t_ln1_g + l * kD, t_ln1_b + l * kD, hbuf);
    gemm_kernel<false, true, false, false><<<dim3(M / 64, 3 * kD / 128), blk256, 0, stream>>>(
        hbuf, t_wqkv + (size_t)l * kD * 3 * kD, t_bqkv + (size_t)l * 3 * kD,
        nullptr, qkv, M, 3 * kD, kD);
    attn_kernel<1><<<attn_grid, blk128, ATTN_LDS, stream>>>(
        qkv, 3 * kD, qkv + kD, 3 * kD, qkv + 2 * kD, 3 * kD, tgt_ids, ctx, kD);
    gemm_kernel<false, true, true, false><<<dim3(M / 64, kD / 128), blk256, 0, stream>>>(
        ctx, t_wo1 + (size_t)l * kD * kD, t_bo1 + (size_t)l * kD, xt, xt, M, kD, kD);

    ln_kernel<<<M, blk256, 0, stream>>>(xt, t_ln2_g + l * kD, t_ln2_b + l * kD, hbuf);
    gemm_kernel<false, true, false, false><<<dim3(M / 64, kD / 128), blk256, 0, stream>>>(
        hbuf, t_wq + (size_t)l * kD * kD, t_bq + (size_t)l * kD, nullptr, qb, M, kD, kD);
    gemm_kernel<false, true, false, false><<<dim3(M / 64, 2 * kD / 128), blk256, 0, stream>>>(
        memf, t_wkv + (size_t)l * kD * 2 * kD, t_bkv + (size_t)l * 2 * kD,
        nullptr, qkv, M, 2 * kD, kD);
    attn_kernel<0><<<attn_grid, blk128, ATTN_LDS, stream>>>(
        qb, kD, qkv, 2 * kD, qkv + kD, 2 * kD, src_ids, ctx, kD);
    gemm_kernel<false, true, true, false><<<dim3(M / 64, kD / 128), blk256, 0, stream>>>(
        ctx, t_wo2 + (size_t)l * kD * kD, t_bo2 + (size_t)l * kD, xt, xt, M, kD, kD);

    ln_kernel<<<M, blk256, 0, stream>>>(xt, t_ln3_g + l * kD, t_ln3_b + l * kD, hbuf);
    gemm_kernel<false, true, false, true><<<dim3(M / 64, kF / 128), blk256, 0, stream>>>(
        hbuf, t_w1 + (size_t)l * kD * kF, t_b1 + (size_t)l * kF, nullptr, mid, M, kF, kD);
    gemm_kernel<false, true, true, false><<<dim3(M / 64, kD / 128), blk256, 0, stream>>>(
        mid, t_w2 + (size_t)l * kF * kD, t_b2 + (size_t)l * kD, xt, xt, M, kD, kF);
  }
  ln_kernel<<<M, blk256, 0, stream>>>(xt, df_g, df_b, hbuf);

  // ---- tied output projection: logits = h @ tgt_emb^T ----
  gemm_kernel<true, false, false, false><<<dim3(M / 64, kV / 128), blk256, 0, stream>>>(
      hbuf, tgt_emb, nullptr, nullptr, (float*)d_out, M, kV, kD);
}